// TopkLayer_noCLStopk_71640054497836
// MI455X (gfx1250) — compile-verified
//
#include <hip/hip_runtime.h>
#include <hip/hip_bf16.h>

// Problem geometry (from reference): x (32, 4097, 768) fp32, top-k along the
// 4096 non-CLS tokens per (batch, feature) column, k = int(4096*0.1) = 409.
#define B_N      32
#define T_TOT    4097
#define HW       4096
#define D_FULL   768
#define D_TILE   16
#define KSEL     409
#define NTHREADS 512
#define NWAVES   (NTHREADS / 32)          // 16 waves
#define T_SLICE  (HW / NWAVES)            // 256 rows per wave TDM descriptor
#define TGN      (NTHREADS / D_TILE)      // 32 t-groups in scan loops

#define SMEM_DATA_BYTES  (HW * D_TILE * 4)   // 262144: staged tile, data[t][dl]
#define SMEM_HIST_BYTES  (D_TILE * 256 * 4)  // 16384: per-column radix histograms
#define SMEM_MISC_BYTES  (4 * D_TILE * 4)    // pref/kleft/tiecnt + pad
#define SMEM_BYTES (SMEM_DATA_BYTES + SMEM_HIST_BYTES + SMEM_MISC_BYTES)

#ifndef __has_builtin
#define __has_builtin(x) 0
#endif

#if __has_builtin(__builtin_amdgcn_tensor_load_to_lds) && \
    __has_builtin(__builtin_amdgcn_tensor_store_from_lds) && \
    __has_builtin(__builtin_amdgcn_s_wait_tensorcnt)
#define USE_TDM 1
#else
#define USE_TDM 0
#endif

typedef __attribute__((ext_vector_type(4))) unsigned int v4u;
typedef __attribute__((ext_vector_type(8))) int          v8i;
typedef __attribute__((ext_vector_type(4))) int          v4i;

#if USE_TDM
// Build a 2D TDM descriptor: tile = D_TILE contiguous floats (dim0) x
// T_SLICE rows (dim1), row stride = D_FULL floats. Groups 2/3 zero (<=2D).
__device__ __forceinline__ void tdm_make_desc(unsigned long long gbyte, unsigned ldsaddr,
                                              v4u& g0, v8i& g1) {
    g0.x = 1u;                                   // count=1 (valid user descriptor)
    g0.y = ldsaddr;                              // lds_addr (bytes)
    g0.z = (unsigned)(gbyte & 0xFFFFFFFFull);    // global_addr[31:0]
    g0.w = (unsigned)((gbyte >> 32) & 0x1FFFFFFull) | (2u << 30); // addr[56:32] | type=2
    g1[0] = (int)(2u << 16);                     // data_size = 4 bytes
    g1[1] = (int)((unsigned)D_FULL << 16);       // tensor_dim0[15:0] @ [31:16]
    g1[2] = (int)((unsigned)T_SLICE << 16);      // tensor_dim0 hi=0 | tensor_dim1 lo
    g1[3] = (int)((unsigned)D_TILE << 16);       // tensor_dim1 hi=0 | tile_dim0
    g1[4] = (int)(unsigned)T_SLICE;              // tile_dim1 | tile_dim2=0
    g1[5] = (int)(unsigned)D_FULL;               // tensor_dim0_stride lo32
    g1[6] = 0;                                   // stride hi | dim1_stride lo
    g1[7] = 0;
}
#endif

__global__ void __launch_bounds__(NTHREADS, 1)
TopkLayer_noCLStopk_71640054497836_kernel(const float* __restrict__ x,
                                          float* __restrict__ out) {
    extern __shared__ unsigned char smem_raw[];
    float*    data   = (float*)smem_raw;
    unsigned* hist   = (unsigned*)(smem_raw + SMEM_DATA_BYTES);
    unsigned* pref   = (unsigned*)(smem_raw + SMEM_DATA_BYTES + SMEM_HIST_BYTES);
    unsigned* kleft  = pref + D_TILE;
    unsigned* tiecnt = kleft + D_TILE;

    const unsigned tid = threadIdx.x;
    const unsigned b   = blockIdx.y;
    const unsigned d0  = blockIdx.x * D_TILE;

    // element offset of (b, t=1, d0): start of this WG's maskable region
    const unsigned long long base =
        ((unsigned long long)b * T_TOT + 1ull) * D_FULL + d0;

    // CLS row (t=0) passthrough for this WG's 16 features.
    if (tid < D_TILE) {
        unsigned long long o = (unsigned long long)b * T_TOT * D_FULL + d0 + tid;
        out[o] = x[o];
        pref[tid]   = 0u;
        kleft[tid]  = KSEL;
        tiecnt[tid] = 0u;
    }

    // ---- Stage the (4096 x 16) strided tile into LDS via the TDM ----
#if USE_TDM
    {
        const int wv = __builtin_amdgcn_readfirstlane((int)(tid >> 5)); // wave id, SGPR
        const unsigned long long gbyte =
            (unsigned long long)(size_t)x +
            (base + (unsigned long long)wv * T_SLICE * D_FULL) * 4ull;
        const unsigned ldsaddr =
            __builtin_amdgcn_groupstaticsize() + (unsigned)wv * (T_SLICE * D_TILE * 4);
        v4u g0; v8i g1; v4i gz4 = {0, 0, 0, 0}; v8i gz8 = {0, 0, 0, 0, 0, 0, 0, 0};
        tdm_make_desc(gbyte, ldsaddr, g0, g1);
        __builtin_amdgcn_tensor_load_to_lds(g0, g1, gz4, gz4, gz8, 0);
        __builtin_amdgcn_s_wait_tensorcnt(0);
    }
#else
    for (unsigned i = tid; i < HW * D_TILE; i += NTHREADS) {
        unsigned t = i / D_TILE, dl = i % D_TILE;
        data[i] = x[base + (unsigned long long)t * D_FULL + dl];
    }
#endif
    __syncthreads();

    // ---- Exact 4x8-bit radix select on |x| bit patterns, per column ----
    const unsigned dl = tid & (D_TILE - 1);
    const unsigned tg = tid / D_TILE;
    const unsigned shifts[4]  = {23u, 15u, 7u, 0u};
    const unsigned premask[4] = {0u, 0x7F800000u, 0x7FFF8000u, 0x7FFFFF80u};

    for (int lvl = 0; lvl < 4; ++lvl) {
        for (unsigned i = tid; i < D_TILE * 256; i += NTHREADS) hist[i] = 0u;
        __syncthreads();
        const unsigned pv = pref[dl];
        const unsigned pm = premask[lvl];
        const unsigned sh = shifts[lvl];
        for (unsigned t = tg; t < HW; t += TGN) {
            unsigned u = __float_as_uint(data[t * D_TILE + dl]) & 0x7FFFFFFFu;
            if ((u & pm) == (pv & pm)) {
                unsigned bin = (u >> sh) & 0xFFu;
                atomicAdd(&hist[dl * 256 + bin], 1u);
            }
        }
        __syncthreads();
        if (tid < D_TILE) {
            unsigned need = kleft[tid];
            unsigned cum = 0, binSel = 0;
            for (int bn = 255; bn >= 0; --bn) {
                unsigned h = hist[tid * 256 + (unsigned)bn];
                if (cum + h >= need) { binSel = (unsigned)bn; kleft[tid] = need - cum; break; }
                cum += h;
            }
            pref[tid] |= (binSel << sh);
        }
        __syncthreads();
    }

    // pref[dl] = exact k-th largest |x| bit pattern; kleft[dl] = #equals to keep
    const unsigned Tthr = pref[dl];
    const unsigned rEq  = kleft[dl];

    // ---- Apply mask (exactly K survivors per column) ----
    for (unsigned t = tg; t < HW; t += TGN) {
        float v = data[t * D_TILE + dl];
        unsigned u = __float_as_uint(v) & 0x7FFFFFFFu;
        float o;
        if (u > Tthr) {
            o = v;
        } else if (u == Tthr) {
            unsigned old = atomicAdd(&tiecnt[dl], 1u);
            o = (old < rEq) ? v : 0.0f;
        } else {
            o = 0.0f;
        }
#if USE_TDM
        data[t * D_TILE + dl] = o;     // stage masked tile for TDM store
#else
        out[base + (unsigned long long)t * D_FULL + dl] = o;
#endif
    }

#if USE_TDM
    __syncthreads();
    {
        const int wv = __builtin_amdgcn_readfirstlane((int)(tid >> 5));
        const unsigned long long gbyte =
            (unsigned long long)(size_t)out +
            (base + (unsigned long long)wv * T_SLICE * D_FULL) * 4ull;
        const unsigned ldsaddr =
            __builtin_amdgcn_groupstaticsize() + (unsigned)wv * (T_SLICE * D_TILE * 4);
        v4u g0; v8i g1; v4i gz4 = {0, 0, 0, 0}; v8i gz8 = {0, 0, 0, 0, 0, 0, 0, 0};
        tdm_make_desc(gbyte, ldsaddr, g0, g1);
        __builtin_amdgcn_tensor_store_from_lds(g0, g1, gz4, gz4, gz8, 0);
        __builtin_amdgcn_s_wait_tensorcnt(0);
    }
#endif
}

extern "C" void kernel_launch(void* const* d_in, const int* in_sizes, int n_in,
                              void* d_out, int out_size, void* d_ws, size_t ws_size,
                              hipStream_t stream) {
    const float* x = (const float*)d_in[0];
    float* out = (float*)d_out;
    (void)in_sizes; (void)n_in; (void)d_ws; (void)ws_size; (void)out_size;

    // Allow the large dynamic-LDS allocation (host-side attribute set; not a
    // stream op, safe under graph capture).
    (void)hipFuncSetAttribute(
        reinterpret_cast<const void*>(&TopkLayer_noCLStopk_71640054497836_kernel),
        hipFuncAttributeMaxDynamicSharedMemorySize, (int)SMEM_BYTES);

    dim3 grid(D_FULL / D_TILE, B_N);   // 48 x 32 = 1536 workgroups
    dim3 block(NTHREADS);
    TopkLayer_noCLStopk_71640054497836_kernel<<<grid, block, SMEM_BYTES, stream>>>(x, out);
}